// CorticalGrid_89172111000129
// MI455X (gfx1250) — compile-verified
//
#include <hip/hip_runtime.h>
#include <math.h>

// ---------------------------------------------------------------------------
// CorticalGrid predictive-coding iteration on MI455X (gfx1250).
//   - One workgroup (512 threads = 16 wave32) per grid column.
//   - fp32 WMMA (v_wmma_f32_16x16x4_f32) for all four per-column GEMMs.
//   - One kernel launch per step (grid-wide neighbor dependency), ping-pong
//     state buffers in the workspace; final step writes x_obj into d_out.
// ---------------------------------------------------------------------------

#define GRID_HH 32
#define GRID_WW 32
#define OBJ_DIM 64
#define LOC_DIM 16
#define SDIM    64          // 8x8 sensory patch
#define NCOLS   (GRID_HH * GRID_WW)
#define BATCH   64
#define ETA_X   0.05f
#define NSTEPS  20

#define XO_TOT  ((size_t)NCOLS * BATCH * OBJ_DIM)   // 4,194,304
#define XL_TOT  ((size_t)NCOLS * BATCH * LOC_DIM)   // 1,048,576

typedef __attribute__((ext_vector_type(2))) float v2f;
typedef __attribute__((ext_vector_type(8))) float v8f;

__device__ __forceinline__ v8f wmma_f32_k4(v2f a, v2f b, v8f c) {
  // 8 args: (neg_a, A, neg_b, B, c_mod, C, reuse_a, reuse_b)
  return __builtin_amdgcn_wmma_f32_16x16x4_f32(
      false, a, false, b, (short)0, c, false, false);
}

// Zero the initial state (x_obj0, x_loc0) and the 20-entry energy history.
__global__ void cg_init_kernel(float* __restrict__ xo,
                               float* __restrict__ xl,
                               float* __restrict__ energy) {
  size_t i = (size_t)blockIdx.x * blockDim.x + threadIdx.x;
  size_t stride = (size_t)gridDim.x * blockDim.x;
  for (size_t e = i; e < XO_TOT; e += stride) xo[e] = 0.0f;
  for (size_t e = i; e < XL_TOT; e += stride) xl[e] = 0.0f;
  if (i < NSTEPS) energy[i] = 0.0f;
}

// One relaxation step for all 1024 columns.
__global__ __launch_bounds__(512)
void cg_step_kernel(const float* __restrict__ gin,     // (B, 65536)
                    const float* __restrict__ Wobj,    // (N, 64, 64)
                    const float* __restrict__ Wloc,    // (N, 16, 64)
                    const float* __restrict__ xo_in,   // (N, B, 64)
                    const float* __restrict__ xl_in,   // (N, B, 16)
                    float* __restrict__ xo_out,        // (N, B, 64)
                    float* __restrict__ xl_out,        // (N, B, 16)
                    float* __restrict__ energy_t) {    // scalar accumulator
  // Row-padded LDS tiles (stride 65 / 17 -> conflict-free column walks).
  __shared__ float sWo[OBJ_DIM][65];   // W_obj (O x S)
  __shared__ float sWl[LOC_DIM][65];   // W_loc (L x S)
  __shared__ float sPG[BATCH][65];     // patch, overwritten in-place by g
  __shared__ float sXo[BATCH][65];     // x_obj (B x O)
  __shared__ float sXl[BATCH][17];     // x_loc (B x L)
  __shared__ float sCtx[BATCH][65];    // neighbour-mean context (B x O)
  __shared__ float sEnergy;

  const int n   = blockIdx.x;
  const int gh  = n >> 5;
  const int gw  = n & 31;
  const int tid = threadIdx.x;

  if (tid == 0) sEnergy = 0.0f;

  // ---- Phase A: stage operands in LDS -----------------------------------
  for (int e = tid; e < OBJ_DIM * SDIM; e += 512)
    sWo[e >> 6][e & 63] = Wobj[(size_t)n * OBJ_DIM * SDIM + e];
  for (int e = tid; e < LOC_DIM * SDIM; e += 512)
    sWl[e >> 6][e & 63] = Wloc[(size_t)n * LOC_DIM * SDIM + e];

  // Gather the 8x8 patch for this column from the global image.
  for (int e = tid; e < BATCH * SDIM; e += 512) {
    int b = e >> 6, s = e & 63;
    int py = s >> 3, px = s & 7;
    sPG[b][s] = gin[(size_t)b * 65536 + (size_t)(gh * 8 + py) * 256 + gw * 8 + px];
  }

  // x_obj + lateral context (mean over valid 4-neighbours of previous state).
  const float inv_cnt =
      1.0f / (float)((gh > 0) + (gh < GRID_HH - 1) + (gw > 0) + (gw < GRID_WW - 1));
  for (int e = tid; e < BATCH * OBJ_DIM; e += 512) {
    sXo[e >> 6][e & 63] = xo_in[(size_t)n * BATCH * OBJ_DIM + e];
    float ssum = 0.0f;
    if (gh > 0)           ssum += xo_in[(size_t)(n - GRID_WW) * BATCH * OBJ_DIM + e];
    if (gh < GRID_HH - 1) ssum += xo_in[(size_t)(n + GRID_WW) * BATCH * OBJ_DIM + e];
    if (gw > 0)           ssum += xo_in[(size_t)(n - 1) * BATCH * OBJ_DIM + e];
    if (gw < GRID_WW - 1) ssum += xo_in[(size_t)(n + 1) * BATCH * OBJ_DIM + e];
    sCtx[e >> 6][e & 63] = ssum * inv_cnt;
  }
  for (int e = tid; e < BATCH * LOC_DIM; e += 512)
    sXl[e >> 4][e & 15] = xl_in[(size_t)n * BATCH * LOC_DIM + e];
  __syncthreads();

  // ---- Phase B: forward U = Xo*Wo + Xl*Wl, g = (P-tanh(U))*(1-tanh^2) ---
  const int lane = tid & 31;
  const int wv   = tid >> 5;          // 0..15 : one 16x16 tile per wave
  const int kh   = lane >> 4;         // half-wave select
  const int mr   = lane & 15;
  const int mt   = (wv >> 2) * 16;    // batch-tile base
  const int nt   = (wv & 3) * 16;     // sensory/obj-tile base

  v8f acc = {};
  for (int k = 0; k < SDIM; k += 4) {
    v2f a, b;
    a.x = sXo[mt + mr][k + 2 * kh];
    a.y = sXo[mt + mr][k + 2 * kh + 1];
    b.x = sWo[k + 2 * kh][nt + mr];
    b.y = sWo[k + 2 * kh + 1][nt + mr];
    acc = wmma_f32_k4(a, b, acc);
  }
  for (int k = 0; k < LOC_DIM; k += 4) {
    v2f a, b;
    a.x = sXl[mt + mr][k + 2 * kh];
    a.y = sXl[mt + mr][k + 2 * kh + 1];
    b.x = sWl[k + 2 * kh][nt + mr];
    b.y = sWl[k + 2 * kh + 1][nt + mr];
    acc = wmma_f32_k4(a, b, acc);
  }

  float epart = 0.0f;
#pragma unroll
  for (int r = 0; r < 8; ++r) {
    int bb = mt + r + 8 * kh;         // C/D layout: M = r + 8*(lane>=16)
    int ss = nt + mr;                 //             N = lane & 15
    float pred = tanhf(acc[r]);
    float eps  = sPG[bb][ss] - pred;
    float g    = eps * (1.0f - pred * pred);
    epart += 0.5f * eps * eps;
    sPG[bb][ss] = g;                  // in-place: each element owned by one lane
  }

  // Wave-level energy reduction, then one LDS accumulate per wave.
  for (int off = 16; off > 0; off >>= 1) epart += __shfl_xor(epart, off, 32);
  if (lane == 0) atomicAdd(&sEnergy, epart);
  __syncthreads();                    // sG complete + sEnergy complete

  // ---- Phase C: backward dXo = G*Wo^T (+ctx-x), dXl = G*Wl^T ------------
  v8f acc2 = {};
  for (int k = 0; k < SDIM; k += 4) {
    v2f a, b;
    a.x = sPG[mt + mr][k + 2 * kh];
    a.y = sPG[mt + mr][k + 2 * kh + 1];
    b.x = sWo[nt + mr][k + 2 * kh];       // Wo^T[k][o] = Wo[o][k]
    b.y = sWo[nt + mr][k + 2 * kh + 1];
    acc2 = wmma_f32_k4(a, b, acc2);
  }
#pragma unroll
  for (int r = 0; r < 8; ++r) {
    int bb = mt + r + 8 * kh;
    int oo = nt + mr;
    float xo = sXo[bb][oo];
    xo_out[(size_t)n * BATCH * OBJ_DIM + bb * OBJ_DIM + oo] =
        xo + ETA_X * (acc2[r] + sCtx[bb][oo] - xo);
  }

  // dXl: 4 tiles; all waves compute (keeps EXEC all-ones for WMMA),
  // only waves 0..3 store.
  const int mt2 = (wv & 3) * 16;
  v8f acc3 = {};
  for (int k = 0; k < SDIM; k += 4) {
    v2f a, b;
    a.x = sPG[mt2 + mr][k + 2 * kh];
    a.y = sPG[mt2 + mr][k + 2 * kh + 1];
    b.x = sWl[mr][k + 2 * kh];            // Wl^T[k][l] = Wl[l][k]
    b.y = sWl[mr][k + 2 * kh + 1];
    acc3 = wmma_f32_k4(a, b, acc3);
  }
  if (wv < 4) {
#pragma unroll
    for (int r = 0; r < 8; ++r) {
      int bb = mt2 + r + 8 * kh;
      int ll = mr;
      float xl = sXl[bb][ll];
      xl_out[(size_t)n * BATCH * LOC_DIM + bb * LOC_DIM + ll] =
          xl + ETA_X * acc3[r];
    }
  }

  if (tid == 0) atomicAdd(energy_t, sEnergy);
}

extern "C" void kernel_launch(void* const* d_in, const int* in_sizes, int n_in,
                              void* d_out, int out_size, void* d_ws, size_t ws_size,
                              hipStream_t stream) {
  const float* gin  = (const float*)d_in[0];   // (64, 65536) f32
  const float* Wobj = (const float*)d_in[1];   // (1024, 64, 64) f32
  const float* Wloc = (const float*)d_in[2];   // (1024, 16, 64) f32
  // d_in[3] = steps (== 20 per reference setup)

  float* out    = (float*)d_out;               // [x_obj (4194304) | energy (20)]
  float* energy = out + XO_TOT;

  float* ws  = (float*)d_ws;                   // ping-pong state: ~42 MB
  float* xoA = ws;
  float* xoB = xoA + XO_TOT;
  float* xlA = xoB + XO_TOT;
  float* xlB = xlA + XL_TOT;

  cg_init_kernel<<<4096, 256, 0, stream>>>(xoA, xlA, energy);

  for (int t = 0; t < NSTEPS; ++t) {
    const float* xo_in = (t & 1) ? xoB : xoA;
    const float* xl_in = (t & 1) ? xlB : xlA;
    float* xo_out = (t == NSTEPS - 1) ? out : ((t & 1) ? xoA : xoB);
    float* xl_out = (t & 1) ? xlA : xlB;
    cg_step_kernel<<<NCOLS, 512, 0, stream>>>(gin, Wobj, Wloc,
                                              xo_in, xl_in, xo_out, xl_out,
                                              energy + t);
  }
}